// SubGraphNet_83958020702805
// MI455X (gfx1250) — compile-verified
//
#include <hip/hip_runtime.h>

// ---------------------------------------------------------------------------
// SubGraphNet on gfx1250:
//  - concat-GEMM factorization (node pre-GEMMs + fused edge kernel)
//  - fp32 WMMA 16x16x4 for all GEMMs
//  - weight matrices staged into LDS via the Tensor Data Mover
//    (tensor_load_to_lds + s_wait_tensorcnt), K-loop reads LDS branch-free
//  - float atomic scatter-add into agg (L2-resident, 20 MB)
// ---------------------------------------------------------------------------

typedef float        v2f  __attribute__((ext_vector_type(2)));
typedef float        v8f  __attribute__((ext_vector_type(8)));
typedef unsigned int u32;
typedef u32          v4u  __attribute__((ext_vector_type(4)));
typedef int          v8i  __attribute__((ext_vector_type(8)));
typedef int          v4i  __attribute__((ext_vector_type(4)));

#define NNODES 50000
#define NEDGES 800000
#define FDIM   100
#define DDIM   64
#define HEDGE  (NEDGES / 2)

__device__ __forceinline__ v8f wmma_f32(v2f a, v2f b, v8f c) {
  return __builtin_amdgcn_wmma_f32_16x16x4_f32(
      /*neg_a=*/false, a, /*neg_b=*/false, b,
      /*c_mod=*/(short)0, c, /*reuse_a=*/false, /*reuse_b=*/false);
}

// TDM: load a 2D tile (rows x width fp32, row stride stride_elems) from global
// into LDS at byte offset lds_off.  D# per CDNA5 ISA.
// 6-arg builtin form (clang-23 / therock headers):
//   (v4u g0, v8i g1, v4i g2, v4i g3, v8i g4, i32 cpol)
__device__ __forceinline__ void tdm_load_2d(const float* gptr, u32 lds_off,
                                            u32 width, u32 rows,
                                            u32 stride_elems) {
  unsigned long long ga = (unsigned long long)(size_t)gptr;
  v4u g0;
  g0[0] = 1u;                                              // count=1, user D#
  g0[1] = lds_off;                                         // lds_addr (bytes)
  g0[2] = (u32)ga;                                         // global_addr lo
  g0[3] = ((u32)(ga >> 32) & 0x01FFFFFFu) | 0x80000000u;   // addr hi | type=2
  v8i g1;
  g1[0] = (int)(2u << 16);                                 // data_size = 4B
  g1[1] = (int)((width & 0xFFFFu) << 16);                  // tensor_dim0 lo16
  g1[2] = (int)(((width >> 16) & 0xFFFFu) |                // tensor_dim0 hi16
                ((rows & 0xFFFFu) << 16));                 // tensor_dim1 lo16
  g1[3] = (int)(((rows >> 16) & 0xFFFFu) |                 // tensor_dim1 hi16
                ((width & 0xFFFFu) << 16));                // tile_dim0 = width
  g1[4] = (int)(rows & 0xFFFFu);                           // tile_dim1; dim2=0
  g1[5] = (int)stride_elems;                               // dim0_stride lo32
  g1[6] = 0;                                               // stride hi / d1s lo
  g1[7] = 0;
  v4i z4 = {0, 0, 0, 0};
  v8i z8 = {0, 0, 0, 0, 0, 0, 0, 0};
  __builtin_amdgcn_tensor_load_to_lds(g0, g1, z4, z4, z8, 0);
}

// ---------------------------------------------------------------------------
// C[M x Nout] = A[M x K] @ B[K x Nout] (+bias)(+relu), row-major.
// Block: 256 threads, 64x64 tile; whole B (<=100x100) TDM-staged into LDS.
// ---------------------------------------------------------------------------
__global__ __launch_bounds__(256) void gemm_kernel(
    const float* __restrict__ A, const float* __restrict__ B,
    const float* __restrict__ bias, float* __restrict__ C,
    int M, int K, int Nout, int relu)
{
  __shared__ float sB[FDIM * FDIM];          // up to 100x100 = 40 KB

  if (threadIdx.x < 32) {                    // wave 0 issues the TDM load
    tdm_load_2d(B, (u32)(size_t)&sB[0], (u32)Nout, (u32)K, (u32)Nout);
    __builtin_amdgcn_s_wait_tensorcnt(0);
  }
  __syncthreads();

  const int lane = threadIdx.x & 31;
  const int wave = threadIdx.x >> 5;
  const int nidx = lane & 15;
  const int half = lane >> 4;
  const int wm = wave & 3;
  const int wn = wave >> 2;
  const int row0 = blockIdx.x * 64 + wm * 16;
  const int col0 = blockIdx.y * 64 + wn * 32;

  int arow = row0 + nidx;
  if (arow >= M) arow = M - 1;               // clamp; store is guarded
  const int c0 = col0 + nidx;
  const int c1 = col0 + 16 + nidx;
  // Branch-free OOB handling: clamped column + 0/1 mask (k-invariant).
  const int   c0c = (c0 < Nout) ? c0 : (Nout - 1);
  const int   c1c = (c1 < Nout) ? c1 : (Nout - 1);
  const float m0  = (c0 < Nout) ? 1.f : 0.f;
  const float m1  = (c1 < Nout) ? 1.f : 0.f;

  v8f acc0 = {}, acc1 = {};
  for (int k0 = 0; k0 < K; k0 += 4) {        // K % 4 == 0 (100 or 64)
    const int ka = k0 + half * 2;
    v2f a, b0, b1;
    a.x  = A[(size_t)arow * K + ka];
    a.y  = A[(size_t)arow * K + ka + 1];
    b0.x = sB[ka * Nout + c0c] * m0;
    b0.y = sB[(ka + 1) * Nout + c0c] * m0;
    b1.x = sB[ka * Nout + c1c] * m1;
    b1.y = sB[(ka + 1) * Nout + c1c] * m1;
    acc0 = wmma_f32(a, b0, acc0);
    acc1 = wmma_f32(a, b1, acc1);
  }

  #pragma unroll
  for (int r = 0; r < 8; ++r) {
    const int m = row0 + half * 8 + r;
    if (m >= M) continue;
    if (c0 < Nout) {
      float v = acc0[r] + (bias ? bias[c0] : 0.f);
      if (relu) v = fmaxf(v, 0.f);
      C[(size_t)m * Nout + c0] = v;
    }
    if (c1 < Nout) {
      float v = acc1[r] + (bias ? bias[c1] : 0.f);
      if (relu) v = fmaxf(v, 0.f);
      C[(size_t)m * Nout + c1] = v;
    }
  }
}

// ---------------------------------------------------------------------------
// Fused edge-message kernel for one edge half (HEDGE % 128 == 0).
//   msg = relu( (ea @ Wmid) + GA[idxA] + GB[idxB] + bias );  agg[idxD] += msg
// Wmid (100x100) TDM-staged into LDS; 128 edges per block, 16 per wave.
// ---------------------------------------------------------------------------
__global__ __launch_bounds__(256) void edge_kernel(
    const float* __restrict__ ea,      // [HEDGE x F] base of this half
    const int*   __restrict__ idxA,
    const int*   __restrict__ idxB,
    const int*   __restrict__ idxD,
    const float* __restrict__ GA,      // [N x F]
    const float* __restrict__ GB,      // [N x F]
    const float* __restrict__ Wmid,    // [F x F]
    const float* __restrict__ bias,    // [F]
    float*       __restrict__ agg)     // [N x F]
{
  __shared__ float sW[FDIM * FDIM];    // 40 KB
  __shared__ int   sA[128], sB2[128], sD[128];

  const int tid  = threadIdx.x;
  const int eblk = blockIdx.x * 128;
  if (tid < 128) {
    sA[tid]  = idxA[eblk + tid];
    sB2[tid] = idxB[eblk + tid];
    sD[tid]  = idxD[eblk + tid];
  }
  if (tid < 32) {                      // wave 0 issues the TDM load of Wmid
    tdm_load_2d(Wmid, (u32)(size_t)&sW[0], FDIM, FDIM, FDIM);
    __builtin_amdgcn_s_wait_tensorcnt(0);
  }
  __syncthreads();

  const int lane = tid & 31;
  const int wave = tid >> 5;
  const int nidx = lane & 15;
  const int half = lane >> 4;
  const int erow = eblk + wave * 16 + nidx;

  // Per-col-tile clamped column + 0/1 mask (only tile 6 is ever masked).
  int   cc[7];
  float cm[7];
  #pragma unroll
  for (int t = 0; t < 7; ++t) {
    const int c = t * 16 + nidx;
    cc[t] = (c < FDIM) ? c : (FDIM - 1);
    cm[t] = (c < FDIM) ? 1.f : 0.f;
  }

  v8f acc[7] = {};
  for (int k0 = 0; k0 < FDIM; k0 += 4) {
    const int ka = k0 + half * 2;
    v2f a;
    a.x = ea[(size_t)erow * FDIM + ka];
    a.y = ea[(size_t)erow * FDIM + ka + 1];
    #pragma unroll
    for (int t = 0; t < 7; ++t) {
      v2f b;
      b.x = sW[ka * FDIM + cc[t]] * cm[t];
      b.y = sW[(ka + 1) * FDIM + cc[t]] * cm[t];
      acc[t] = wmma_f32(a, b, acc[t]);
    }
  }

  #pragma unroll
  for (int t = 0; t < 7; ++t) {
    const int col = t * 16 + nidx;
    if (col >= FDIM) continue;
    const float bv = bias[col];
    #pragma unroll
    for (int r = 0; r < 8; ++r) {
      const int el = wave * 16 + half * 8 + r;
      float v = acc[t][r]
              + GA[(size_t)sA[el] * FDIM + col]
              + GB[(size_t)sB2[el] * FDIM + col] + bv;
      v = fmaxf(v, 0.f);
      unsafeAtomicAdd(&agg[(size_t)sD[el] * FDIM + col], v);
    }
  }
}

// ---------------------------------------------------------------------------
__global__ void copy_kernel(const float* __restrict__ src,
                            float* __restrict__ dst, size_t n) {
  size_t i = (size_t)blockIdx.x * blockDim.x + threadIdx.x;
  const size_t stride = (size_t)gridDim.x * blockDim.x;
  for (; i < n; i += stride) dst[i] = src[i];
}

__global__ void zero_kernel(float* __restrict__ p, int n) {
  const int i = blockIdx.x * blockDim.x + threadIdx.x;
  if (i < n) p[i] = 0.f;
}

__global__ __launch_bounds__(256) void reduce_kernel(
    const float* __restrict__ X, float* __restrict__ out, int M) {
  const int d    = threadIdx.x & (DDIM - 1);
  const int rgrp = threadIdx.x / DDIM;          // 0..3
  float s = 0.f;
  for (int r = blockIdx.x * 4 + rgrp; r < M; r += gridDim.x * 4)
    s += X[(size_t)r * DDIM + d];
  unsafeAtomicAdd(&out[d], s);
}

// ---------------------------------------------------------------------------
extern "C" void kernel_launch(void* const* d_in, const int* in_sizes, int n_in,
                              void* d_out, int out_size, void* d_ws, size_t ws_size,
                              hipStream_t stream) {
  (void)in_sizes; (void)n_in; (void)out_size; (void)ws_size;
  const int N = NNODES, E = NEDGES, F = FDIM, D = DDIM, H = HEDGE;
  const size_t NF = (size_t)N * F;

  float* ws    = (float*)d_ws;            // 7 * N*F floats = 140 MB
  float* x_cur = ws;
  float* agg   = ws + 1 * NF;
  float* hbuf  = ws + 2 * NF;
  float* P     = ws + 3 * NF;
  float* Q     = ws + 4 * NF;
  float* R     = ws + 5 * NF;
  float* S     = ws + 6 * NF;

  float* out = (float*)d_out;
  zero_kernel<<<1, 128, 0, stream>>>(out, 2 * D);

  auto gemm = [&](const float* A, const float* B, const float* bias, float* C,
                  int M, int K, int Nout, bool relu) {
    dim3 grid((M + 63) / 64, (Nout + 63) / 64);
    gemm_kernel<<<grid, 256, 0, stream>>>(A, B, bias, C, M, K, Nout, relu ? 1 : 0);
  };

  for (int g = 0; g < 2; ++g) {
    const float* x_in = (const float*)d_in[g ? 3 : 0];
    const int*   ei   = (const int*)  d_in[g ? 4 : 1];
    const float* ea   = (const float*)d_in[g ? 5 : 2];
    const int* src = ei;
    const int* dst = ei + E;

    const float* X = x_in;
    for (int layer = 0; layer < 2; ++layer) {
      const float* lw  = (const float*)d_in[layer ? 14 : 6];
      const float* lb  = (const float*)d_in[layer ? 15 : 7];
      const float* l2w = (const float*)d_in[layer ? 16 : 8];
      const float* l2b = (const float*)d_in[layer ? 17 : 9];
      const float* w1  = (const float*)d_in[layer ? 18 : 10];
      const float* b1  = (const float*)d_in[layer ? 19 : 11];
      const float* w2  = (const float*)d_in[layer ? 20 : 12];
      const float* b2  = (const float*)d_in[layer ? 21 : 13];
      const int  Dh       = layer ? D : F;
      const bool relu_out = (layer == 0);

      gemm(X, lw,              nullptr, P, N, F, F, false);  // x @ lw[0:F]
      gemm(X, lw  + 2 * F * F, nullptr, Q, N, F, F, false);  // x @ lw[2F:3F]
      gemm(X, l2w,             nullptr, R, N, F, F, false);  // x @ l2w[0:F]
      gemm(X, l2w + 2 * F * F, nullptr, S, N, F, F, false);  // x @ l2w[2F:3F]

      copy_kernel<<<1024, 256, 0, stream>>>(X, agg, NF);     // agg = x

      edge_kernel<<<H / 128, 256, 0, stream>>>(
          ea, dst, src, dst, P, Q, lw + F * F, lb, agg);
      edge_kernel<<<H / 128, 256, 0, stream>>>(
          ea + (size_t)H * F, src + H, dst + H, dst + H, R, S, l2w + F * F, l2b, agg);

      gemm(agg,  w1, b1, hbuf,  N, F,  Dh, true);
      gemm(hbuf, w2, b2, x_cur, N, Dh, Dh, relu_out);
      X = x_cur;
    }

    reduce_kernel<<<256, 256, 0, stream>>>(x_cur, out + g * D, N);
  }
}